// ImprovedLDConv_25262997635334
// MI455X (gfx1250) — compile-verified
//
#include <hip/hip_runtime.h>
#include <math.h>

#define Bn   4
#define Cn   64
#define Hn   128
#define Wn   128
#define On   64
#define K2n  9
#define CKn  576     // Cn * K2n
#define HWn  16384   // Hn * Wn
#define ST   40      // im2col LDS row stride (dwords); 2*40 mod 64 = 16 -> halves hit disjoint banks

typedef __attribute__((ext_vector_type(2))) float v2f;
typedef __attribute__((ext_vector_type(8))) float v8f;

// ---------------- Kernel 1: offset conv (3x3, pad 1) ----------------
__global__ __launch_bounds__(256) void ldc_offset_conv(
    const float* __restrict__ x, const float* __restrict__ ow,
    const float* __restrict__ ob, float* __restrict__ off_out)
{
    __shared__ float wlds[18 * CKn];           // 41.5 KB
    const int t = threadIdx.x;
    for (int e = t; e < 18 * CKn; e += 256) wlds[e] = ow[e];
    __syncthreads();

    const int gid = blockIdx.x * 256 + t;      // exact grid: B*H*W/256 blocks
    const int b   = gid / HWn;
    const int pix = gid % HWn;
    const int h = pix >> 7, w = pix & 127;

    float acc[18];
#pragma unroll
    for (int oc = 0; oc < 18; ++oc) acc[oc] = 0.f;

    const float* xb = x + b * Cn * HWn;
    for (int c = 0; c < Cn; ++c) {
        const float* xc = xb + c * HWn;
#pragma unroll
        for (int ky = 0; ky < 3; ++ky) {
            const int y = h - 1 + ky;
            const bool yok = (y >= 0) && (y < Hn);
#pragma unroll
            for (int kx = 0; kx < 3; ++kx) {
                const int xw = w - 1 + kx;
                const float v = (yok && xw >= 0 && xw < Wn) ? xc[y * Wn + xw] : 0.f;
                const int widx = c * 9 + ky * 3 + kx;
#pragma unroll
                for (int oc = 0; oc < 18; ++oc)
                    acc[oc] = fmaf(wlds[oc * CKn + widx], v, acc[oc]);
            }
        }
    }
    float* op = off_out + b * 18 * HWn + pix;
#pragma unroll
    for (int oc = 0; oc < 18; ++oc) op[oc * HWn] = acc[oc] + ob[oc];
}

// ---- Kernel 2: deformable gather (LDS im2col) + fp32 WMMA GEMM ----
// Block: 256 threads = 8 waves; tile = 32 pixels (one quarter row) x all 64 O.
__global__ __launch_bounds__(256) void ldc_deform_wmma(
    const float* __restrict__ x, const float* __restrict__ wgt,
    const float* __restrict__ bias, const float* __restrict__ off,
    float* __restrict__ raw)
{
    __shared__ float smB[288 * ST];            // 46 KB: K-chunk x 32 pixels
    __shared__ int   sy0[288], sx0[288];
    __shared__ float swy[288], swx[288];

    const int t    = threadIdx.x;
    const int blk  = blockIdx.x;
    const int b    = blk >> 9;                 // 512 tiles per image
    const int tile = blk & 511;
    const int row  = tile >> 2;
    const int col0 = (tile & 3) << 5;

    // Stage 1: 9*32 bilinear descriptors (shared across all 64 C)
    for (int e = t; e < 288; e += 256) {
        const int k = e >> 5, p = e & 31;
        const int w = col0 + p;
        const float* ofb = off + (b * 18 + 2 * k) * HWn + row * Wn + w;
        const float dy = ofb[0];
        const float dx = ofb[HWn];
        const float py = dy + (float)(row - 1 + (k / 3));
        const float px = dx + (float)(w   - 1 + (k % 3));
        const float fy = floorf(py), fx = floorf(px);
        sy0[e] = (int)fy;  sx0[e] = (int)fx;
        swy[e] = py - fy;  swx[e] = px - fx;
    }

    const int lane  = t & 31;
    const int wave  = t >> 5;
    const int mtile = wave & 3;                // O-tile (4 x 16 = 64)
    const int ntile = wave >> 2;               // pixel half (2 x 16 = 32)
    const int koff  = (lane & 16) ? 2 : 0;     // upper half-wave holds K=2,3
    const int ln    = lane & 15;

    const int g = t >> 5;                      // fill: c-group 0..7
    const int p = t & 31;                      // fill: pixel (consecutive lanes -> coalesced x)

    // Dual accumulators: break the WMMA C->D serial dependency chain
    v8f acc0 = {}, acc1 = {};

    for (int chunk = 0; chunk < 2; ++chunk) {
        __syncthreads();                        // desc RAW (first) / smB WAR (later)
        const int cbase = chunk * 32;
        const float* xb = x + b * Cn * HWn;
#pragma unroll
        for (int i = 0; i < 4; ++i) {
            const int c = cbase + g + 8 * i;
            const float* xc = xb + c * HWn;
#pragma unroll
            for (int k = 0; k < 9; ++k) {
                const int e = k * 32 + p;
                const int y0 = sy0[e], x0 = sx0[e];
                const float wy1 = swy[e], wx1 = swx[e];
                const float wy0 = 1.f - wy1, wx0 = 1.f - wx1;
                const int y1 = y0 + 1, x1 = x0 + 1;
                const bool y0ok = (y0 >= 0) && (y0 < Hn), y1ok = (y1 >= 0) && (y1 < Hn);
                const bool x0ok = (x0 >= 0) && (x0 < Wn), x1ok = (x1 >= 0) && (x1 < Wn);
                const float g00 = (y0ok && x0ok) ? xc[y0 * Wn + x0] : 0.f;
                const float g01 = (y0ok && x1ok) ? xc[y0 * Wn + x1] : 0.f;
                const float g10 = (y1ok && x0ok) ? xc[y1 * Wn + x0] : 0.f;
                const float g11 = (y1ok && x1ok) ? xc[y1 * Wn + x1] : 0.f;
                const float v = wy0 * (wx0 * g00 + wx1 * g01)
                              + wy1 * (wx0 * g10 + wx1 * g11);
                smB[((g + 8 * i) * 9 + k) * ST + p] = v;
            }
        }
        __syncthreads();

        // GEMM: 72 x v_wmma_f32_16x16x4_f32 per wave per chunk, 2-way interleaved
        const int   o    = mtile * 16 + ln;
        const float* wr  = wgt + o * CKn + cbase * 9;   // L2-resident weights
        const int   pixn = ntile * 16 + ln;
#pragma unroll 2
        for (int k0 = 0; k0 < 288; k0 += 8) {
            v2f a0, b0, a1, b1;
            a0[0] = wr[k0 + koff];
            a0[1] = wr[k0 + koff + 1];
            a1[0] = wr[k0 + 4 + koff];
            a1[1] = wr[k0 + 4 + koff + 1];
            b0[0] = smB[(k0 + koff) * ST + pixn];
            b0[1] = smB[(k0 + koff + 1) * ST + pixn];
            b1[0] = smB[(k0 + 4 + koff) * ST + pixn];
            b1[1] = smB[(k0 + 4 + koff + 1) * ST + pixn];
            acc0 = __builtin_amdgcn_wmma_f32_16x16x4_f32(
                false, a0, false, b0, (short)0, acc0, false, false);
            acc1 = __builtin_amdgcn_wmma_f32_16x16x4_f32(
                false, a1, false, b1, (short)0, acc1, false, false);
        }
    }

    // Epilogue: D layout — VGPR r: lanes 0-15 -> M=r, lanes 16-31 -> M=r+8
    const int obase = mtile * 16 + ((lane & 16) ? 8 : 0);
    float* rb = raw + (b * On) * HWn + row * Wn + col0 + ntile * 16 + ln;
#pragma unroll
    for (int r = 0; r < 8; ++r) {
        const int o = obase + r;
        rb[o * HWn] = acc0[r] + acc1[r] + bias[o];
    }
}

// ---- Kernel 3: per-channel sum/sumsq partials (deterministic, no atomics) ----
// float4 loads -> global_load_b128: fewer instructions per byte on the 23.3 TB/s path
__global__ __launch_bounds__(256) void ldc_reduce(
    const float* __restrict__ raw, float* __restrict__ part)
{
    __shared__ float s1[256], s2[256];
    const int o = blockIdx.x >> 3, seg = blockIdx.x & 7;
    const int t = threadIdx.x;
    float a = 0.f, q = 0.f;
    for (int bb = 0; bb < Bn; ++bb) {
        const float4* base = (const float4*)(raw + (bb * On + o) * HWn + seg * 2048);
#pragma unroll
        for (int j = 0; j < 2; ++j) {           // 512 float4 per segment / 256 threads
            const float4 v = base[t + j * 256];
            a += v.x + v.y + v.z + v.w;
            q += v.x * v.x + v.y * v.y + v.z * v.z + v.w * v.w;
        }
    }
    s1[t] = a; s2[t] = q;
    __syncthreads();
    for (int s = 128; s > 0; s >>= 1) {
        if (t < s) { s1[t] += s1[t + s]; s2[t] += s2[t + s]; }
        __syncthreads();
    }
    if (t == 0) { part[o * 8 + seg] = s1[0]; part[512 + o * 8 + seg] = s2[0]; }
}

// ---- Kernel 4: finalize mean / invstd ----
__global__ void ldc_finalize(const float* __restrict__ part, float* __restrict__ stats)
{
    const int o = threadIdx.x;
    if (o < 64) {
        float s = 0.f, q = 0.f;
        for (int j = 0; j < 8; ++j) { s += part[o * 8 + j]; q += part[512 + o * 8 + j]; }
        const float invN = 1.f / (float)(Bn * HWn);
        const float mean = s * invN;
        const float var  = q * invN - mean * mean;
        stats[o]      = mean;
        stats[64 + o] = rsqrtf(var + 1e-5f);
    }
}

// ---- Kernel 5: BN + SiLU (float4 in / float4 out) ----
__global__ __launch_bounds__(256) void ldc_bn_silu(
    const float* __restrict__ raw, const float* __restrict__ stats,
    const float* __restrict__ gamma, const float* __restrict__ beta,
    float* __restrict__ out)
{
    const int idx4 = blockIdx.x * 256 + threadIdx.x;   // float4 index
    const int o = (idx4 >> 12) & 63;                   // 4 consecutive floats share a channel
    const float mean = stats[o], istd = stats[64 + o];
    const float ga = gamma[o], be = beta[o];
    float4 v = ((const float4*)raw)[idx4];
    float r[4] = {v.x, v.y, v.z, v.w};
#pragma unroll
    for (int j = 0; j < 4; ++j) {
        float u = (r[j] - mean) * istd * ga + be;
        r[j] = u / (1.f + __expf(-u));
    }
    float4 ov = {r[0], r[1], r[2], r[3]};
    ((float4*)out)[idx4] = ov;
}

extern "C" void kernel_launch(void* const* d_in, const int* in_sizes, int n_in,
                              void* d_out, int out_size, void* d_ws, size_t ws_size,
                              hipStream_t stream)
{
    const float* x        = (const float*)d_in[0];
    const float* weight   = (const float*)d_in[1];
    const float* bias     = (const float*)d_in[2];
    const float* offset_w = (const float*)d_in[3];
    const float* offset_b = (const float*)d_in[4];
    const float* gamma    = (const float*)d_in[5];
    const float* beta     = (const float*)d_in[6];
    float* out = (float*)d_out;

    float* ws       = (float*)d_ws;
    float* ws_off   = ws;                             // B*18*H*W   = 1,179,648 f32
    float* ws_raw   = ws_off + Bn * 18 * HWn;         // B*64*H*W   = 4,194,304 f32
    float* ws_part  = ws_raw + Bn * On * HWn;         // 1024 f32
    float* ws_stats = ws_part + 1024;                 // 128 f32

    ldc_offset_conv<<<(Bn * HWn) / 256, 256, 0, stream>>>(x, offset_w, offset_b, ws_off);
    ldc_deform_wmma<<<Bn * 512, 256, 0, stream>>>(x, weight, bias, ws_off, ws_raw);
    ldc_reduce<<<512, 256, 0, stream>>>(ws_raw, ws_part);
    ldc_finalize<<<1, 64, 0, stream>>>(ws_part, ws_stats);
    ldc_bn_silu<<<(Bn * On * HWn) / (4 * 256), 256, 0, stream>>>(ws_raw, ws_stats, gamma, beta, out);
}